// Morphology_84602265797171
// MI455X (gfx1250) — compile-verified
//
#include <hip/hip_runtime.h>
#include <cstdint>
#include <cstddef>

// Morphological dilation2d (max-plus conv), B=4, Cin=4, Cout=4, H=W=1024, k=5, pad=2.
// CDNA5 path: async global->LDS tile copy (ASYNCcnt) + LDS-tiled VALU max-plus.
// o is split across wave-groups (readfirstlane -> SGPR) to keep weight loads scalar
// with only 25 live weights per c-iteration: no SGPR spills, ~56 VGPRs.

#define TW      64                  // tile width  (pixels)
#define TH      16                  // tile height (pixels)
#define HALO    2
#define LW      (TW + 2 * HALO)     // 68 used cols in LDS
#define LH      (TH + 2 * HALO)     // 20 rows in LDS
#define LSTRIDE 72                  // padded row stride (floats), 288B (16B-aligned)
#define C_IN    4
#define C_OUT   4
#define KK      5
#define PPT     16                  // pixels per thread (horizontal run)
#define IMG     1024

typedef __attribute__((ext_vector_type(4))) float v4f;

// Async copy one dword global -> LDS (CDNA5 GLOBAL_LOAD_ASYNC_TO_LDS_B32, ASYNCcnt).
__device__ __forceinline__ void async_g2l_b32(uint32_t lds_off, const float* gp) {
    asm volatile("global_load_async_to_lds_b32 %0, %1, off"
                 :: "v"(lds_off), "v"((uint64_t)(uintptr_t)gp)
                 : "memory");
}

__device__ __forceinline__ void wait_asynccnt0() {
    asm volatile("s_wait_asynccnt 0x0" ::: "memory");
}

__global__ __launch_bounds__(256)
void Morphology_84602265797171_kernel(const float* __restrict__ x,
                                      const float* __restrict__ w,
                                      float* __restrict__ out) {
    __shared__ float lds[C_IN * LH * LSTRIDE];   // 4*20*72*4 = 23040 bytes

    const int b   = blockIdx.z;
    const int th0 = blockIdx.y * TH;             // tile origin (h)
    const int tw0 = blockIdx.x * TW;             // tile origin (w)
    const int tid = threadIdx.x;

    // ---------------- Stage 1: fill LDS tile (4 channels + zero halo) ----------------
    const int NEL = C_IN * LH * LW;              // 5440 elements
    for (int idx = tid; idx < NEL; idx += 256) {
        const int cc = idx % LW;
        const int t  = idx / LW;
        const int r  = t % LH;
        const int c  = t / LH;
        const int gh = th0 + r  - HALO;
        const int gw = tw0 + cc - HALO;
        float* lp = &lds[(c * LH + r) * LSTRIDE + cc];
        if ((unsigned)gh < (unsigned)IMG && (unsigned)gw < (unsigned)IMG) {
            const float* gp = x + (((size_t)(b * C_IN + c) * IMG + gh) * IMG + gw);
            async_g2l_b32((uint32_t)(uintptr_t)lp, gp);
        } else {
            *lp = 0.0f;                          // zero padding (ds_store, DScnt)
        }
    }
    wait_asynccnt0();        // all async global->LDS transfers complete
    __syncthreads();         // LDS visible workgroup-wide

    // ---------------- Stage 2: max-plus stencil ----------------
    // 4 groups of 64 threads; group g computes output channel o = g for the whole tile.
    // readfirstlane pins o in an SGPR so weight loads stay scalar (s_load).
    const int o   = __builtin_amdgcn_readfirstlane(tid >> 6);  // 0..3, wave-uniform
    const int ln  = tid & 63;
    const int row = ln & 15;                     // 0..15
    const int col = (ln >> 4) * PPT;             // 0, 16, 32, 48 (16B-aligned)

    float acc[PPT];
#pragma unroll
    for (int p = 0; p < PPT; ++p)
        acc[p] = -3.402823466e38f;

#pragma unroll 1
    for (int c = 0; c < C_IN; ++c) {
#pragma unroll
        for (int i = 0; i < KK; ++i) {
            const float* lrow = &lds[(c * LH + (row + i)) * LSTRIDE + col];
            float xr[PPT + 4];
#pragma unroll
            for (int q = 0; q < 5; ++q) {        // 5x ds_load_b128 (20-float window)
                const v4f t = *(const v4f*)(lrow + 4 * q);
                xr[4 * q + 0] = t[0];
                xr[4 * q + 1] = t[1];
                xr[4 * q + 2] = t[2];
                xr[4 * q + 3] = t[3];
            }
#pragma unroll
            for (int j = 0; j < KK; ++j) {
                const float wv = w[((o * C_IN + c) * KK + i) * KK + j]; // s_load
#pragma unroll
                for (int p = 0; p < PPT; ++p)
                    acc[p] = fmaxf(acc[p], xr[j + p] + wv);  // add + max3-fusable
            }
        }
    }

    // ---------------- Stage 3: store 16 pixels for this thread's o ----------------
    const int gh = th0 + row;
    const int gw = tw0 + col;
    float* op = out + (((size_t)(b * C_OUT + o) * IMG + gh) * IMG + gw);
#pragma unroll
    for (int q = 0; q < 4; ++q) {
        v4f v = {acc[4 * q + 0], acc[4 * q + 1], acc[4 * q + 2], acc[4 * q + 3]};
        *(v4f*)(op + 4 * q) = v;                 // global_store_b128
    }
}

extern "C" void kernel_launch(void* const* d_in, const int* in_sizes, int n_in,
                              void* d_out, int out_size, void* d_ws, size_t ws_size,
                              hipStream_t stream) {
    (void)in_sizes; (void)n_in; (void)d_ws; (void)ws_size; (void)out_size;
    const float* x = (const float*)d_in[0];      // (4,4,1024,1024) f32
    const float* w = (const float*)d_in[1];      // (4,4,5,5) f32
    float* out     = (float*)d_out;              // (4,4,1024,1024) f32

    dim3 grid(IMG / TW, IMG / TH, 4);            // (16, 64, 4)
    dim3 block(256, 1, 1);                       // 8 wave32s
    Morphology_84602265797171_kernel<<<grid, block, 0, stream>>>(x, w, out);
}